// YOLOLayer_27341761806387
// MI455X (gfx1250) — compile-verified
//
#include <hip/hip_runtime.h>

// ---------------- problem constants ----------------
#define BB     16
#define CIN    256
#define FFG    76
#define TT     50
#define AAN    3
#define NCLS   80
#define OC     255            // A * (NCLS+5)
#define NCELL  (FFG*FFG)      // 5776
#define TILE_N 64             // cells per workgroup (4 waves x 16)
#define NTILE  91             // ceil(5776/64)
#define NWG    (NTILE*BB)     // 1456
#define KSTEPS 64             // 256 / 4 (K per WMMA)
#define WROWSTRIDE 260        // padded LDS row stride (dwords) for weights

typedef __attribute__((ext_vector_type(2))) float v2f;
typedef __attribute__((ext_vector_type(8))) float v8f;
typedef __attribute__((ext_vector_type(4))) unsigned int v4u;
typedef __attribute__((ext_vector_type(4))) int v4i;
typedef __attribute__((ext_vector_type(8))) int v8i;

#if defined(__has_builtin)
#if __has_builtin(__builtin_amdgcn_tensor_load_to_lds)
#define USE_TDM 1
#endif
#endif

// ALL_ANCHORS / 8
__constant__ float c_aw[9] = {1.25f, 2.0f, 4.125f, 3.75f, 7.75f, 7.375f, 14.5f, 19.5f, 46.625f};
__constant__ float c_ah[9] = {1.625f, 3.75f, 2.875f, 7.625f, 5.625f, 14.875f, 11.25f, 24.75f, 40.75f};

__device__ __forceinline__ float clogf_(float z) {
  return z > 0.f ? fmaxf(logf(z), -100.f) : -100.f;
}
__device__ __forceinline__ float sigm_(float x) {
  return 1.f / (1.f + expf(-x));
}

// ---------------- kernel 0: init workspace ----------------
__global__ void k_init(int* __restrict__ tmap, int* __restrict__ n_t, int ntm) {
  int idx = blockIdx.x * blockDim.x + threadIdx.x;
  if (idx < ntm) tmap[idx] = -1;
  if (idx < BB)  n_t[idx] = 0;
}

// ---------------- kernel 1: per-target preprocessing ----------------
__global__ void k_targets(const float* __restrict__ targets,
                          float* __restrict__ gtb, float* __restrict__ validf,
                          float* __restrict__ trec, int* __restrict__ tmap,
                          int* __restrict__ n_t) {
  int idx = blockIdx.x * blockDim.x + threadIdx.x;
  if (idx >= BB * TT) return;
  int b = idx / TT, t = idx % TT;
  const float* tg = targets + idx * 5;
  float t0 = tg[0], t1 = tg[1], t2 = tg[2], t3 = tg[3], t4 = tg[4];
  bool valid = (t0 + t1 + t2 + t3 + t4) > 0.f;
  float gx = t0 * FFG, gy = t1 * FFG, gw = t2 * FFG, gh = t3 * FFG;
  gtb[idx*4+0] = gx; gtb[idx*4+1] = gy; gtb[idx*4+2] = gw; gtb[idx*4+3] = gh;
  validf[idx] = valid ? 1.f : 0.f;
  if (valid) atomicAdd(&n_t[b], 1);
  // best of 9 anchors by IoU(wh), first-wins argmax
  float best = -1e30f; int bi = 0;
  for (int j = 0; j < 9; ++j) {
    float inter = fminf(gw, c_aw[j]) * fminf(gh, c_ah[j]);
    float uni = gw*gh + c_aw[j]*c_ah[j] - inter;
    float r = inter / (uni + 1e-16f);
    if (r > best) { best = r; bi = j; }
  }
  bool keep = valid && (bi < AAN);
  if (keep) {
    int gxi = (int)gx, gyi = (int)gy;
    if (gxi >= 0 && gxi < FFG && gyi >= 0 && gyi < FFG) {   // mode='drop'
      tmap[((b*AAN + bi)*FFG + gyi)*FFG + gxi] = t;
      float* rec = trec + idx*6;
      rec[0] = gx - floorf(gx);
      rec[1] = gy - floorf(gy);
      rec[2] = logf(gw / c_aw[bi] + 1e-16f);
      rec[3] = logf(gh / c_ah[bi] + 1e-16f);
      rec[4] = sqrtf(2.f - gw*gh / (float)(FFG*FFG));
      rec[5] = t4;
    }
  }
}

// ---------------- kernel 2: fused WMMA GEMM + loss ----------------
// LDS layout (floats): wlds[64*260] | outl[256*64] | gtbs[256] | red[128]
#define SM_WLDS 0
#define SM_OUTL (64*WROWSTRIDE)
#define SM_GTBS (SM_OUTL + 256*64)
#define SM_RED  (SM_GTBS + 256)
#define SM_FLOATS (SM_RED + 128)

__global__ __launch_bounds__(128)
void k_main(const float* __restrict__ x, const float* __restrict__ conv_w,
            const float* __restrict__ conv_b, const float* __restrict__ gtb,
            const float* __restrict__ validf, const float* __restrict__ trec,
            const int* __restrict__ tmap, const int* __restrict__ n_t,
            float* __restrict__ part) {
  extern __shared__ float smem[];
  float* wlds = smem + SM_WLDS;
  float* outl = smem + SM_OUTL;
  float* gtbs = smem + SM_GTBS;
  float* red  = smem + SM_RED;

  const int tid  = threadIdx.x;
  const int b    = blockIdx.y;
  const int n0   = blockIdx.x * TILE_N;
  const int lane = tid & 31;
  const int wv   = tid >> 5;      // wave 0..3 -> 16-cell N-subtile
  const int col  = lane & 15;     // N (and A-row M) index within tile
  const int hi   = lane >> 4;     // half-wave selects K pair / M+8

  // stage GT boxes + valid flags for this batch into LDS
  if (tid < TT) {
    const float* g = gtb + (b*TT + tid)*4;
    gtbs[tid*5+0] = g[0]; gtbs[tid*5+1] = g[1];
    gtbs[tid*5+2] = g[2]; gtbs[tid*5+3] = g[3];
    gtbs[tid*5+4] = validf[b*TT + tid];
  }

  // ---- load B fragments (x) for this wave's 16 cells, all 64 K-steps ----
  // B 4x16 layout: VGPR j holds K = 4k + j + 2*hi, N = col
  const int ncell = n0 + wv*16 + col;
  const int nc = ncell < NCELL ? ncell : (NCELL-1);     // clamp (EXEC stays full)
  const float* xb = x + (size_t)b * CIN * NCELL + nc;
  v2f bfrag[KSTEPS];
#pragma unroll
  for (int k = 0; k < KSTEPS; ++k) {
    int c = 4*k + 2*hi;
    bfrag[k].x = xb[(size_t)c * NCELL];
    bfrag[k].y = xb[(size_t)(c+1) * NCELL];
  }

  // ---- GEMM: 16 M-tiles in 4 chunks of 64 weight rows through LDS ----
  for (int chunk = 0; chunk < 4; ++chunk) {
    __syncthreads();
#ifdef USE_TDM
    // Tensor Data Mover: DMA 64 weight rows (256 dwords each) into LDS,
    // hardware-padding 4 dwords after every 256 (-> 260-dword stride) and
    // zero-filling reads past tensor_dim1 (handles the o>=255 pad row).
    if (wv == 0) {                       // TDM ignores EXEC; issue from one wave
      unsigned long long ga =
          (unsigned long long)(const void*)(conv_w + (size_t)chunk * 64 * CIN);
      int rows_left = OC - chunk * 64;   // 255,191,127,63
      v4u g0;
      g0.x = 1u;                                   // count=1, user descriptor
      g0.y = (unsigned)(SM_WLDS * 4);              // lds_addr (bytes)
      g0.z = (unsigned)(ga & 0xFFFFFFFFu);         // global_addr[31:0]
      g0.w = (unsigned)((ga >> 32) & 0x1FFFFFFu)   // global_addr[56:32]
             | (2u << 30);                         // type=2 ("image")
      v8i g1;
      g1[0] = (2 << 16)                            // data_size=4B
            | (1 << 20)                            // pad_enable
            | (7 << 22)                            // pad_interval: 256 dwords
            | (3 << 25);                           // pad_amount: 4 dwords
      g1[1] = (CIN & 0xFFFF) << 16;                // tensor_dim0[15:0]=256
      g1[2] = (rows_left & 0xFFFF) << 16;          // tensor_dim1[15:0]
      g1[3] = (CIN & 0xFFFF) << 16;                // tile_dim0=256
      g1[4] = 64;                                  // tile_dim1=64, tile_dim2=0
      g1[5] = CIN;                                 // tensor_dim0_stride=256
      g1[6] = 0;
      g1[7] = 0;
      v4i gz = {0, 0, 0, 0};                       // 2-D tensor: groups 2/3 unused
#if __clang_major__ >= 23
      v8i gz8 = {0, 0, 0, 0, 0, 0, 0, 0};
      __builtin_amdgcn_tensor_load_to_lds(g0, g1, gz, gz, gz8, 0);
#else
      __builtin_amdgcn_tensor_load_to_lds(g0, g1, gz, gz, 0);
#endif
      __builtin_amdgcn_s_wait_tensorcnt(0);
    }
#else
    // fallback: cooperative load, 64 rows x 256 cols, padded stride 260 dwords
    for (int i = tid; i < 64*64; i += 128) {
      int rr = i >> 6, c4 = i & 63;
      int og = chunk*64 + rr;
      float4 v = make_float4(0.f, 0.f, 0.f, 0.f);
      if (og < OC) v = *(const float4*)(conv_w + (size_t)og*CIN + c4*4);
      *(float4*)(wlds + rr*WROWSTRIDE + c4*4) = v;
    }
#endif
    __syncthreads();
#pragma unroll
    for (int mm = 0; mm < 4; ++mm) {
      v8f acc = {};
      const float* arow = wlds + (mm*16 + col)*WROWSTRIDE + 2*hi;
#pragma unroll
      for (int k = 0; k < KSTEPS; ++k) {
        v2f a = *(const v2f*)(arow + 4*k);   // A: M=col, K pair = 4k+2hi..+1
        acc = __builtin_amdgcn_wmma_f32_16x16x4_f32(
            false, a, false, bfrag[k], (short)0, acc, false, false);
      }
      int m = chunk*4 + mm;
#pragma unroll
      for (int r = 0; r < 8; ++r) {
        int o = m*16 + r + 8*hi;             // C/D: VGPR r -> M=r (+8 for hi half)
        float bias = (o < OC) ? conv_b[o] : 0.f;
        outl[o*64 + wv*16 + col] = acc[r] + bias;
      }
    }
  }
  __syncthreads();

  // ---- fused loss over (anchor, cell) ----
  float s_xy = 0.f, s_wh = 0.f, s_obj = 0.f, s_cls = 0.f;
  const int nt = n_t[b];
  for (int i = tid; i < AAN * TILE_N; i += 128) {
    int a = i >> 6, cl = i & 63;
    int n = n0 + cl;
    if (n >= NCELL) continue;
    int gy = n / FFG, gx = n - gy*FFG;
    int o0 = a * (NCLS + 5);
    float ox = outl[(o0+0)*64 + cl];
    float oy = outl[(o0+1)*64 + cl];
    float ow = outl[(o0+2)*64 + cl];
    float oh = outl[(o0+3)*64 + cl];
    float oo = outl[(o0+4)*64 + cl];
    float sx = sigm_(ox), sy = sigm_(oy);
    float px = sx + (float)gx, py = sy + (float)gy;
    float pw = expf(ow) * c_aw[a], ph = expf(oh) * c_ah[a];
    // max IoU vs all (valid) GT boxes of this batch
    float maxiou = 0.f;
    for (int t = 0; t < TT; ++t) {
      float gxc = gtbs[t*5+0], gyc = gtbs[t*5+1];
      float gw  = gtbs[t*5+2], gh  = gtbs[t*5+3];
      float vfl = gtbs[t*5+4];
      float lox = fmaxf(px - pw*0.5f, gxc - gw*0.5f);
      float hix = fminf(px + pw*0.5f, gxc + gw*0.5f);
      float loy = fmaxf(py - ph*0.5f, gyc - gh*0.5f);
      float hiy = fminf(py + ph*0.5f, gyc + gh*0.5f);
      float ia  = fmaxf(hix-lox, 0.f) * fmaxf(hiy-loy, 0.f);
      float iou = ia / (pw*ph + gw*gh - ia + 1e-16f);
      maxiou = fmaxf(maxiou, iou * vfl);
    }
    float high = (maxiou > 0.7f) ? 1.f : 0.f;
    float om = (a < nt) ? (1.f - high) : 1.f;
    int ti = tmap[((b*AAN + a)*FFG + gy)*FFG + gx];
    if (ti >= 0) om = 1.f;
    float tobj = (ti >= 0) ? 1.f : 0.f;
    // objectness BCE: p = sig*om, t = tobj*om (om in {0,1} -> om=0 contributes 0)
    float po = sigm_(oo) * om, to = tobj * om;
    s_obj -= to * clogf_(po) + (1.f - to) * clogf_(1.f - po);
    if (ti >= 0) {
      const float* rec = trec + (b*TT + ti)*6;
      float sc = rec[4], w2 = sc*sc;
      s_xy -= w2 * (rec[0]*clogf_(sx) + (1.f-rec[0])*clogf_(1.f-sx));
      s_xy -= w2 * (rec[1]*clogf_(sy) + (1.f-rec[1])*clogf_(1.f-sy));
      float dw = (ow - rec[2]) * sc, dh = (oh - rec[3]) * sc;
      s_wh += 0.5f * (dw*dw + dh*dh);
      int cls = (int)rec[5];
      for (int c = 0; c < NCLS; ++c) {
        float pc = sigm_(outl[(o0+5+c)*64 + cl]);
        float tc = (c == cls) ? 1.f : 0.f;
        s_cls -= tc * clogf_(pc) + (1.f - tc) * clogf_(1.f - pc);
      }
    }
  }

  // ---- deterministic block reduction of 4 components ----
  float comps[4] = {s_xy, s_wh, s_obj, s_cls};
  int wg = b * gridDim.x + blockIdx.x;
  for (int c = 0; c < 4; ++c) {
    __syncthreads();
    red[tid] = comps[c];
    __syncthreads();
    for (int st = 64; st > 0; st >>= 1) {
      if (tid < st) red[tid] += red[tid + st];
      __syncthreads();
    }
    if (tid == 0) part[wg*4 + c] = red[0];
  }
}

// ---------------- kernel 3: fixed-order final reduction ----------------
__global__ __launch_bounds__(256)
void k_reduce(const float* __restrict__ part, float* __restrict__ out) {
  __shared__ float red[256];
  int tid = threadIdx.x;
  float keep[4];
  for (int c = 0; c < 4; ++c) {
    float s = 0.f;
    for (int i = tid; i < NWG; i += 256) s += part[i*4 + c];
    red[tid] = s;
    __syncthreads();
    for (int st = 128; st > 0; st >>= 1) {
      if (tid < st) red[tid] += red[tid + st];
      __syncthreads();
    }
    keep[c] = red[0];
    __syncthreads();
  }
  if (tid == 0) {
    out[1] = keep[0]; out[2] = keep[1]; out[3] = keep[2]; out[4] = keep[3];
    out[0] = keep[0] + keep[1] + keep[2] + keep[3];
  }
}

// ---------------- launch ----------------
extern "C" void kernel_launch(void* const* d_in, const int* in_sizes, int n_in,
                              void* d_out, int out_size, void* d_ws, size_t ws_size,
                              hipStream_t stream) {
  const float* x       = (const float*)d_in[0];
  const float* targets = (const float*)d_in[1];
  const float* conv_w  = (const float*)d_in[2];
  const float* conv_b  = (const float*)d_in[3];
  float* out = (float*)d_out;

  // workspace carve
  char* ws = (char*)d_ws;
  const int TMAP_N = BB * AAN * NCELL;                  // 277248
  int*   tmap   = (int*)ws;
  int*   n_t    = (int*)(ws + (size_t)TMAP_N * 4);
  float* gtb    = (float*)(ws + (size_t)TMAP_N * 4 + 64);
  float* validf = gtb + BB*TT*4;
  float* trec   = validf + BB*TT;
  float* part   = trec + BB*TT*6;

  k_init<<<(TMAP_N + 255)/256, 256, 0, stream>>>(tmap, n_t, TMAP_N);
  k_targets<<<(BB*TT + 255)/256, 256, 0, stream>>>(targets, gtb, validf, trec, tmap, n_t);
  dim3 grid(NTILE, BB);
  k_main<<<grid, 128, SM_FLOATS * sizeof(float), stream>>>(
      x, conv_w, conv_b, gtb, validf, trec, tmap, n_t, part);
  k_reduce<<<1, 256, 0, stream>>>(part, out);
}